// TransitionDownFPS_63479616634981
// MI455X (gfx1250) — compile-verified
//
#include <hip/hip_runtime.h>
#include <hip/hip_bf16.h>

// ---------------- problem constants (from reference setup) ----------------
#define NB      2          // batches
#define NPTS    16384      // points per batch
#define MSAMP   4096       // FPS samples per batch (NPTS/4)
#define QTOT    (NB * MSAMP)   // 8192 total queries
#define KNN_K   16
#define CIN     64
#define COUT    128
#define CF      67         // 3 + CIN
#define CFP     68         // padded to multiple of 4
#define NP_ELEMS (QTOT * 3)          // 24576
#define XOUT_ELEMS (QTOT * COUT)     // 1048576
#define NO_OFF  (NP_ELEMS + XOUT_ELEMS)

#define FPS_T   1024
#define KNN_T   256
#define KNN_CHUNK 2048
#define GEMM_T  256        // 8 waves, 1 query per wave

typedef __attribute__((ext_vector_type(2))) float v2f;
typedef __attribute__((ext_vector_type(8))) float v8f;

// =====================================================================
// Kernel 1: farthest point sampling. One block per batch, 1024 threads.
// Each thread owns 16 points fully in registers (coords + min-dist).
// Exact first-index argmax tie-break to match jnp.argmax.
// =====================================================================
__global__ __launch_bounds__(FPS_T) void fps_kernel(const float* __restrict__ p,
                                                    float* __restrict__ out,
                                                    float* __restrict__ qscr) {
    const int b   = blockIdx.x;
    const int tid = threadIdx.x;
    const float* pb = p + (size_t)b * NPTS * 3;

    float px[16], py[16], pz[16], dd[16];
#pragma unroll
    for (int i = 0; i < 16; ++i) {
        int j = i * FPS_T + tid;
        px[i] = pb[3 * j + 0];
        py[i] = pb[3 * j + 1];
        pz[i] = pb[3 * j + 2];
        dd[i] = 3.4e38f;
    }

    __shared__ float s_val[32];
    __shared__ int   s_idx[32];
    __shared__ float s_cur[3];

    float cx = pb[0], cy = pb[1], cz = pb[2];
    if (tid == 0) {
        size_t o3 = (size_t)b * MSAMP * 3;
        out[o3 + 0] = cx; out[o3 + 1] = cy; out[o3 + 2] = cz;
        qscr[o3 + 0] = cx; qscr[o3 + 1] = cy; qscr[o3 + 2] = cz;
        if (b == 0) {  // n_o = [M, 2M] as float in the concatenated output
            out[NO_OFF + 0] = (float)MSAMP;
            out[NO_OFF + 1] = (float)(2 * MSAMP);
        }
    }

    const int lane = tid & 31;
    const int wid  = tid >> 5;

    for (int it = 1; it < MSAMP; ++it) {
        float bv = -1.0f;
        int   bj = 0;
#pragma unroll
        for (int i = 0; i < 16; ++i) {
            float dx = px[i] - cx, dy = py[i] - cy, dz = pz[i] - cz;
            float d  = dx * dx + dy * dy + dz * dz;
            float dm = fminf(dd[i], d);
            dd[i] = dm;
            // j ascends with i, so strict > keeps the smallest index on ties
            if (dm > bv) { bv = dm; bj = i * FPS_T + tid; }
        }
        // wave32 argmax reduction (smaller index wins ties)
#pragma unroll
        for (int off = 16; off > 0; off >>= 1) {
            float ov = __shfl_down(bv, off, 32);
            int   oj = __shfl_down(bj, off, 32);
            if (ov > bv || (ov == bv && oj < bj)) { bv = ov; bj = oj; }
        }
        if (lane == 0) { s_val[wid] = bv; s_idx[wid] = bj; }
        __syncthreads();
        if (wid == 0) {
            float v = s_val[lane];
            int   j = s_idx[lane];
#pragma unroll
            for (int off = 16; off > 0; off >>= 1) {
                float ov = __shfl_down(v, off, 32);
                int   oj = __shfl_down(j, off, 32);
                if (ov > v || (ov == v && oj < j)) { v = ov; j = oj; }
            }
            if (lane == 0) {
                float nx = pb[3 * j + 0], ny = pb[3 * j + 1], nz = pb[3 * j + 2];
                s_cur[0] = nx; s_cur[1] = ny; s_cur[2] = nz;
                size_t o3 = ((size_t)b * MSAMP + it) * 3;
                out[o3 + 0] = nx; out[o3 + 1] = ny; out[o3 + 2] = nz;
                qscr[o3 + 0] = nx; qscr[o3 + 1] = ny; qscr[o3 + 2] = nz;
            }
        }
        __syncthreads();
        cx = s_cur[0]; cy = s_cur[1]; cz = s_cur[2];
    }
}

// =====================================================================
// Kernel 2: brute-force KNN. One thread per query; points staged in LDS
// chunks; register-resident sorted top-16 via unrolled insertion.
// =====================================================================
__global__ __launch_bounds__(KNN_T) void knn_kernel(const float* __restrict__ p,
                                                    const float* __restrict__ qscr,
                                                    int* __restrict__ knn_idx) {
    __shared__ float sp[KNN_CHUNK * 3];
    const int qid = blockIdx.x * KNN_T + threadIdx.x;   // block never spans batches
    const int b   = qid >> 12;                          // / MSAMP
    const float* pb = p + (size_t)b * NPTS * 3;

    const float qx = qscr[3 * qid + 0];
    const float qy = qscr[3 * qid + 1];
    const float qz = qscr[3 * qid + 2];

    float bd[KNN_K];
    int   bi[KNN_K];
#pragma unroll
    for (int s = 0; s < KNN_K; ++s) { bd[s] = 3.4e38f; bi[s] = 0; }

    for (int c0 = 0; c0 < NPTS; c0 += KNN_CHUNK) {
        __syncthreads();
        for (int t = threadIdx.x; t < KNN_CHUNK * 3; t += KNN_T)
            sp[t] = pb[(size_t)c0 * 3 + t];
        __syncthreads();
        for (int j = 0; j < KNN_CHUNK; ++j) {
            float dx = sp[3 * j + 0] - qx;
            float dy = sp[3 * j + 1] - qy;
            float dz = sp[3 * j + 2] - qz;
            float d2 = dx * dx + dy * dy + dz * dz;
            if (d2 < bd[KNN_K - 1]) {
                float nd = d2;
                int   ni = c0 + j;
#pragma unroll
                for (int s = 0; s < KNN_K; ++s) {
                    if (nd < bd[s]) {
                        float td = bd[s]; bd[s] = nd; nd = td;
                        int   ti = bi[s]; bi[s] = ni; ni = ti;
                    }
                }
            }
        }
    }
#pragma unroll
    for (int s = 0; s < KNN_K; ++s) knn_idx[(size_t)qid * KNN_K + s] = bi[s];
}

// ---------------------------------------------------------------------
__device__ __forceinline__ float featv(int j, float rx, float ry, float rz,
                                       const float* __restrict__ xg) {
    if (j == 0) return rx;
    if (j == 1) return ry;
    if (j == 2) return rz;
    return (j < CF) ? xg[j - 3] : 0.0f;   // col 67 is zero padding
}

// =====================================================================
// Kernel 3: gather + fp32 WMMA GEMM (16x68 @ 68x128) + LayerNorm + ReLU
// + max-pool over the 16 neighbors. One wave (wave32) per query.
//   A layout (16x4 f32): lane L -> row L%16, K-pair 2*(L/16)+{0,1}
//   B layout (4x16 f32): lane L -> col L%16, K-pair 2*(L/16)+{0,1}
//   C/D (16x16 f32):     lane L, vgpr v -> row v+8*(L/16), col L%16
// =====================================================================
__global__ __launch_bounds__(GEMM_T) void group_gemm_ln_pool_kernel(
        const float* __restrict__ p, const float* __restrict__ x,
        const float* __restrict__ W, const float* __restrict__ gamma,
        const float* __restrict__ beta, const float* __restrict__ qscr,
        const int* __restrict__ knn_idx, float* __restrict__ out) {

    __shared__ float sW[CFP * COUT];   // 68x128 fp32 = 34.8 KB (row 67 zeroed)
    __shared__ float sG[COUT];
    __shared__ float sB[COUT];

    for (int t = threadIdx.x; t < CFP * COUT; t += GEMM_T)
        sW[t] = (t < CF * COUT) ? W[t] : 0.0f;
    for (int t = threadIdx.x; t < COUT; t += GEMM_T) { sG[t] = gamma[t]; sB[t] = beta[t]; }
    __syncthreads();

    const int wid  = threadIdx.x >> 5;
    const int lane = threadIdx.x & 31;
    const int q    = blockIdx.x * (GEMM_T / 32) + wid;
    const int b    = q >> 12;                 // / MSAMP
    const int row  = lane & 15;               // neighbor index (A-matrix row)
    const int hi   = lane >> 4;               // half-wave selects K-pair / row block

    const int   nidx = knn_idx[(size_t)q * KNN_K + row];
    const size_t g   = (size_t)b * NPTS + (size_t)nidx;

    const float qx = qscr[3 * q + 0], qy = qscr[3 * q + 1], qz = qscr[3 * q + 2];
    const float rx = p[3 * g + 0] - qx;
    const float ry = p[3 * g + 1] - qy;
    const float rz = p[3 * g + 2] - qz;
    const float* xg = x + g * CIN;

    // per-lane A-matrix fragments: 17 chunks of K=4, this lane holds 2 cols/chunk
    float a0[17], a1[17];
#pragma unroll
    for (int c = 0; c < 17; ++c) {
        int j0 = 4 * c + 2 * hi;
        a0[c] = featv(j0,     rx, ry, rz, xg);
        a1[c] = featv(j0 + 1, rx, ry, rz, xg);
    }

    v8f   accs[8];
    float s1[8], s2[8];
#pragma unroll
    for (int v = 0; v < 8; ++v) { s1[v] = 0.0f; s2[v] = 0.0f; }

#pragma unroll
    for (int t = 0; t < 8; ++t) {           // 8 output-column tiles of 16
        const int col = row + 16 * t;
        v8f acc = {};
#pragma unroll
        for (int c = 0; c < 17; ++c) {      // K = 68 in chunks of 4
            const int k0 = 4 * c + 2 * hi;
            v2f av; av.x = a0[c];                 av.y = a1[c];
            v2f bv; bv.x = sW[k0 * COUT + col];   bv.y = sW[(k0 + 1) * COUT + col];
            acc = __builtin_amdgcn_wmma_f32_16x16x4_f32(
                /*neg_a=*/false, av, /*neg_b=*/false, bv,
                /*c_mod=*/(short)0, acc, /*reuse_a=*/false, /*reuse_b=*/false);
        }
        accs[t] = acc;
#pragma unroll
        for (int v = 0; v < 8; ++v) { float h = acc[v]; s1[v] += h; s2[v] += h * h; }
    }

    // LayerNorm stats per neighbor row: reduce the 16 lanes of each half-wave
    float mean[8], rstd[8];
#pragma unroll
    for (int v = 0; v < 8; ++v) {
        float a = s1[v], c = s2[v];
#pragma unroll
        for (int off = 8; off > 0; off >>= 1) {
            a += __shfl_xor(a, off, 16);
            c += __shfl_xor(c, off, 16);
        }
        float mu = a * (1.0f / (float)COUT);
        float vr = c * (1.0f / (float)COUT) - mu * mu;
        mean[v] = mu;
        rstd[v] = rsqrtf(vr + 1e-5f);
    }

    // normalize + affine + ReLU + max over the 16 neighbors, then store
    const float* xo_base = nullptr; (void)xo_base;
    float* xo = out + NP_ELEMS;
#pragma unroll
    for (int t = 0; t < 8; ++t) {
        const int col = row + 16 * t;
        const float gm = sG[col], bt = sB[col];
        float m = 0.0f;                          // post-ReLU values are >= 0
#pragma unroll
        for (int v = 0; v < 8; ++v) {
            float h = (accs[t][v] - mean[v]) * rstd[v] * gm + bt;
            h = fmaxf(h, 0.0f);
            m = fmaxf(m, h);
        }
        m = fmaxf(m, __shfl_xor(m, 16, 32));     // combine rows 0-7 with 8-15
        if (hi == 0) xo[(size_t)q * COUT + col] = m;
    }
}

// =====================================================================
extern "C" void kernel_launch(void* const* d_in, const int* in_sizes, int n_in,
                              void* d_out, int out_size, void* d_ws, size_t ws_size,
                              hipStream_t stream) {
    const float* p     = (const float*)d_in[0];   // [32768,3]
    const float* x     = (const float*)d_in[1];   // [32768,64]
    /* d_in[2] = o (int32[2]) — offsets are hardcoded by the constants */
    const float* W     = (const float*)d_in[3];   // [67,128]
    const float* gamma = (const float*)d_in[4];   // [128]
    const float* beta  = (const float*)d_in[5];   // [128]
    float* out = (float*)d_out;

    float* qscr = (float*)d_ws;                                   // 8192*3 f32
    int*   knn  = (int*)((char*)d_ws + (size_t)QTOT * 3 * sizeof(float)); // 8192*16 i32

    fps_kernel<<<NB, FPS_T, 0, stream>>>(p, out, qscr);
    knn_kernel<<<QTOT / KNN_T, KNN_T, 0, stream>>>(p, qscr, knn);
    group_gemm_ln_pool_kernel<<<QTOT / (GEMM_T / 32), GEMM_T, 0, stream>>>(
        p, x, W, gamma, beta, qscr, knn, out);
}